// ROIAlign_55018531062382
// MI455X (gfx1250) — compile-verified
//
#include <hip/hip_runtime.h>

// ROI Align forward, MI455X (gfx1250, wave32).
// features: (2,256,200,304) f32 NCHW ; boxes: (R,5) f32 ; out: (R,256,7,7) f32
//
// Memory-bound gather kernel (AI ~3 FLOP/B << fp32 ridge; 0.5 GFLOP vs 175 MB).
// Design:
//  * One workgroup (8 wave32) per ROI: all 49 bins of an ROI execute on one
//    WGP, so the ROI's ~28x28-per-plane corner footprint is reused from WGP$
//    across bins instead of being refetched from L2.
//  * Waves strip-mine bins (bin = waveId, +8); lanes sweep channels, so all
//    bilinear coords/weights are wave-uniform and gather offsets differ only
//    by c*H*W per lane.
//  * Validity is folded into weights -> branch-free, always-in-range gathers.
//  * NT stores for the 50 MB output keep the 124.6 MB feature map resident
//    in the 192 MB L2 (175 MB combined < 192 MB).

#define C_    256
#define H_    200
#define W_    304
#define HW_   (H_ * W_)
#define PH_   7
#define PW_   7
#define BINS_ (PH_ * PW_)
#define SCALE_ 0.25f

// torchvision bilinear_interpolate clamping along one axis (matches _axis_interp).
__device__ __forceinline__ void axis_interp(float coord, int size,
                                            int& lo, int& hi,
                                            float& wl, float& wh) {
    bool valid = (coord >= -1.0f) && (coord <= (float)size);
    float c = fmaxf(coord, 0.0f);
    int l = (int)c;                 // floor for c >= 0
    float fr;
    if (l >= size - 1) {
        lo = size - 1; hi = size - 1; fr = 0.0f;
    } else {
        lo = l; hi = l + 1; fr = c - (float)l;
    }
    wl = valid ? (1.0f - fr) : 0.0f;   // fold validity mask into weights
    wh = valid ? fr : 0.0f;
}

__global__ __launch_bounds__(256) void ROIAlign_55018531062382_kernel(
    const float* __restrict__ feat,
    const float* __restrict__ boxes,
    float* __restrict__ out)
{
    const int r      = blockIdx.x;          // one workgroup per ROI
    const int waveId = threadIdx.x >> 5;
    const int lane   = threadIdx.x & 31;

    // --- per-ROI wave-uniform box math (hoisted out of the bin loop) -------
    const float* box = boxes + (size_t)r * 5;
    const int   b  = (int)box[0];
    const float x1 = box[1] * SCALE_;
    const float y1 = box[2] * SCALE_;
    const float x2 = box[3] * SCALE_;
    const float y2 = box[4] * SCALE_;

    const float roi_w = fmaxf(x2 - x1, 1.0f);
    const float roi_h = fmaxf(y2 - y1, 1.0f);
    const float bin_w = roi_w * (1.0f / (float)PW_);
    const float bin_h = roi_h * (1.0f / (float)PH_);

    const float* fb   = feat + (size_t)b * ((size_t)C_ * HW_);
    float*       obr  = out  + ((size_t)r * C_) * BINS_;

    // --- waves strip-mine the 49 bins of this ROI --------------------------
    for (int bin = waveId; bin < BINS_; bin += 8) {
        const int ph = bin / PW_;
        const int pw = bin - ph * PW_;

        int   yl[2], yh[2], xl[2], xh[2];
        float wyl[2], wyh[2], wxl[2], wxh[2];
#pragma unroll
        for (int s = 0; s < 2; ++s) {
            const float off = ((float)s + 0.5f) * 0.5f;   // (s + 0.5)/G, G=2
            axis_interp(y1 + ((float)ph + off) * bin_h, H_, yl[s], yh[s], wyl[s], wyh[s]);
            axis_interp(x1 + ((float)pw + off) * bin_w, W_, xl[s], xh[s], wxl[s], wxh[s]);
        }

        // --- gather + blend: lanes sweep channels, 2-way unroll for MLP ----
#pragma unroll 2
        for (int i = 0; i < C_ / 32; ++i) {
            const int c = lane + (i << 5);
            const float* plane = fb + (size_t)c * HW_;

            float acc = 0.0f;
#pragma unroll
            for (int sy = 0; sy < 2; ++sy) {
                const float* rl = plane + yl[sy] * W_;
                const float* rh = plane + yh[sy] * W_;
#pragma unroll
                for (int sx = 0; sx < 2; ++sx) {
                    const float a0 = rl[xl[sx]];
                    const float a1 = rl[xh[sx]];
                    const float b0 = rh[xl[sx]];
                    const float b1 = rh[xh[sx]];
                    acc += wyl[sy] * (wxl[sx] * a0 + wxh[sx] * a1)
                         + wyh[sy] * (wxl[sx] * b0 + wxh[sx] * b1);
                }
            }
            // NT store: don't let the output stream evict features from L2.
            __builtin_nontemporal_store(acc * 0.25f, &obr[(size_t)c * BINS_ + bin]);
        }
    }
}

extern "C" void kernel_launch(void* const* d_in, const int* in_sizes, int n_in,
                              void* d_out, int out_size, void* d_ws, size_t ws_size,
                              hipStream_t stream) {
    const float* feat  = (const float*)d_in[0];   // (2,256,200,304) f32
    const float* boxes = (const float*)d_in[1];   // (R,5) f32
    float*       out   = (float*)d_out;           // (R,256,7,7) f32

    const int R = in_sizes[1] / 5;                // one workgroup per ROI
    ROIAlign_55018531062382_kernel<<<R, 256, 0, stream>>>(feat, boxes, out);
}